// SS3D_Seq_SameWeight_alladd_62483184222324
// MI455X (gfx1250) — compile-verified
//
#include <hip/hip_runtime.h>
#include <hip/hip_bf16.h>
#include <math.h>

// ---------------- problem constants ----------------
#define B_SZ    2
#define SP      12           // Dv = Wv = Hv
#define L_SP    1728         // 12^3
#define DM      96           // D_MODEL == D_INNER
#define DST     16           // D_STATE
#define DTR     6            // DT_RANK
#define NDIR    8
#define LT      13824        // 8 * L_SP
#define XDR     38           // DT_RANK + 2*D_STATE

typedef __attribute__((ext_vector_type(2))) float v2f;
typedef __attribute__((ext_vector_type(8))) float v8f;

__device__ __forceinline__ float silu_f(float v) {
    return v / (1.f + __expf(-v));
}
__device__ __forceinline__ float softplus_f(float v) {
    return (v > 20.f) ? v : log1pf(__expf(v));
}

// =====================================================================
// Kernel 1: xz = x @ in_proj_w.T     [3456 x 96] * [96 x 192] -> [3456 x 192]
// One wave per 16x16 tile, V_WMMA_F32_16X16X4_F32, K loop of 24 steps.
// =====================================================================
__global__ __launch_bounds__(128) void k_in_proj(const float* __restrict__ x,
                                                 const float* __restrict__ w,
                                                 float* __restrict__ xz) {
    int wave = (blockIdx.x * blockDim.x + threadIdx.x) >> 5;  // 2592 waves
    int lane = threadIdx.x & 31;
    int mT = wave / 12, nT = wave % 12;     // M tiles: 216, N tiles: 12
    int m0 = mT * 16, n0 = nT * 16;
    int half = lane >> 4, mn = lane & 15;

    const float* arow = x + (size_t)(m0 + mn) * DM;        // A[m, k]
    const float* brow = w + (size_t)(n0 + mn) * DM;        // B[k, n] = W[n, k]

    v8f acc = {};
#pragma unroll
    for (int k0 = 0; k0 < DM; k0 += 4) {
        int k = k0 + 2 * half;
        v2f a; a.x = arow[k]; a.y = arow[k + 1];
        v2f b; b.x = brow[k]; b.y = brow[k + 1];
        acc = __builtin_amdgcn_wmma_f32_16x16x4_f32(false, a, false, b,
                                                    (short)0, acc, false, false);
    }
#pragma unroll
    for (int r = 0; r < 8; ++r)
        xz[(size_t)(m0 + r + 8 * half) * (2 * DM) + n0 + mn] = acc[r];
}

// =====================================================================
// Kernel 2: depthwise 3x3x3 conv (SAME, zero pad) + bias + SiLU
// in : xx = xz[:, 0:96] laid out [pos(3456) x 192]
// out: xc [B, 96, 1728]
// =====================================================================
__global__ __launch_bounds__(256) void k_conv_silu(const float* __restrict__ xz,
                                                   const float* __restrict__ cw,
                                                   const float* __restrict__ cb,
                                                   float* __restrict__ xc) {
    int idx = blockIdx.x * blockDim.x + threadIdx.x;   // 331776 total
    int hv = idx % SP; int t = idx / SP;
    int wv = t % SP;   t /= SP;
    int dv = t % SP;   t /= SP;
    int c  = t % DM;
    int b  = t / DM;

    float acc = cb[c];
#pragma unroll
    for (int kd = 0; kd < 3; ++kd) {
        int dd = dv + kd - 1; if (dd < 0 || dd >= SP) continue;
#pragma unroll
        for (int kw = 0; kw < 3; ++kw) {
            int ww = wv + kw - 1; if (ww < 0 || ww >= SP) continue;
#pragma unroll
            for (int kh = 0; kh < 3; ++kh) {
                int hh = hv + kh - 1; if (hh < 0 || hh >= SP) continue;
                int pos = (dd * SP + ww) * SP + hh;
                acc += cw[((c * 3 + kd) * 3 + kw) * 3 + kh] *
                       xz[(size_t)(b * L_SP + pos) * (2 * DM) + c];
            }
        }
    }
    xc[(size_t)(b * DM + c) * L_SP + (dv * SP + wv) * SP + hv] = silu_f(acc);
}

// =====================================================================
// Kernel 3: build xs [B, 96, 8*1728] = 8 direction orderings of xc
// dirs: (swap34, flip, swap24, flip, swap23, flip, id, flip)
// =====================================================================
__global__ __launch_bounds__(256) void k_dirs(const float* __restrict__ xc,
                                              float* __restrict__ xs) {
    int idx = blockIdx.x * blockDim.x + threadIdx.x;   // 2*96*8*1728
    int l = idx % L_SP; int t = idx / L_SP;
    int dir = t % NDIR; t /= NDIR;
    int d = t % DM;
    int b = t / DM;

    int lf = (dir & 1) ? (L_SP - 1 - l) : l;
    int a = lf / 144, bb = (lf / 12) % 12, c = lf % 12;
    int src;
    switch (dir >> 1) {
        case 0:  src = (a  * SP + c ) * SP + bb; break;  // swapaxes(3,4)
        case 1:  src = (c  * SP + bb) * SP + a;  break;  // swapaxes(2,4)
        case 2:  src = (bb * SP + a ) * SP + c;  break;  // swapaxes(2,3)
        default: src = lf;                       break;  // identity
    }
    xs[(size_t)(b * DM + d) * LT + dir * L_SP + l] =
        xc[(size_t)(b * DM + d) * L_SP + src];
}

// =====================================================================
// Kernel 4: x_dbl[b,c,l] = sum_d x_proj_w[c,d] * xs[b,d,l]
// M = c (38 -> padded 48), N = l (13824), K = 96. WMMA f32 16x16x4.
// =====================================================================
__global__ __launch_bounds__(128) void k_x_proj(const float* __restrict__ xs,
                                                const float* __restrict__ w,
                                                float* __restrict__ xdbl) {
    int wave = (blockIdx.x * blockDim.x + threadIdx.x) >> 5;  // 2*3*864 waves
    int lane = threadIdx.x & 31;
    int nT = wave % (LT / 16); int t = wave / (LT / 16);
    int mT = t % 3; int b = t / 3;
    int m0 = mT * 16, l0 = nT * 16;
    int half = lane >> 4, mn = lane & 15;
    int arow = m0 + mn;                      // padded c row for A

    const float* xsb = xs + (size_t)b * DM * LT + l0 + mn;
    v8f acc = {};
#pragma unroll
    for (int k0 = 0; k0 < DM; k0 += 4) {
        int k = k0 + 2 * half;
        v2f a;
        a.x = (arow < XDR) ? w[arow * DM + k]     : 0.f;
        a.y = (arow < XDR) ? w[arow * DM + k + 1] : 0.f;
        v2f bv;
        bv.x = xsb[(size_t)k * LT];
        bv.y = xsb[(size_t)(k + 1) * LT];
        acc = __builtin_amdgcn_wmma_f32_16x16x4_f32(false, a, false, bv,
                                                    (short)0, acc, false, false);
    }
#pragma unroll
    for (int r = 0; r < 8; ++r) {
        int c = m0 + r + 8 * half;
        if (c < XDR)
            xdbl[(size_t)(b * XDR + c) * LT + l0 + mn] = acc[r];
    }
}

// =====================================================================
// Kernel 5: selective scan, one workgroup per (b,d).
// h[l,n] = exp(delta*A[d,n]) * h[l-1,n] + delta*Bs[n,l]*xs[l]
// delta  = softplus(dt_w[d,:] . x_dbl[:6,l] + dt_b[d])
// y[l]   = sum_n h[l,n]*Cs[n,l] + Ds[d]*xs[l]
// Two-pass chunked scan: 256 threads x 54 elems; LDS scan of (a,b) pairs.
// =====================================================================
#define SCAN_T   256
#define CHUNK    (LT / SCAN_T)   // 54
__global__ __launch_bounds__(SCAN_T) void k_scan(const float* __restrict__ xs,
                                                 const float* __restrict__ xdbl,
                                                 const float* __restrict__ dt_w,
                                                 const float* __restrict__ dt_b,
                                                 const float* __restrict__ A_logs,
                                                 const float* __restrict__ Ds,
                                                 float* __restrict__ y) {
    __shared__ float sA[SCAN_T * DST];
    __shared__ float sB[SCAN_T * DST];

    int bd = blockIdx.x;                 // 192
    int b = bd / DM, d = bd % DM;
    int t = threadIdx.x;

    float An[DST], wdt[DTR];
#pragma unroll
    for (int n = 0; n < DST; ++n) An[n] = -__expf(A_logs[d * DST + n]);
#pragma unroll
    for (int r = 0; r < DTR; ++r) wdt[r] = dt_w[d * DTR + r];
    float dtb = dt_b[d], Dd = Ds[d];

    const float* xsrow = xs + (size_t)(b * DM + d) * LT;
    const float* xdb   = xdbl + (size_t)b * XDR * LT;
    int l0 = t * CHUNK;

    __builtin_prefetch(&xdb[(size_t)(DTR + DST) * LT + l0], 0, 3);  // Cs rows

    // ---- pass 1: per-thread transform composition (aP, bP) ----
    float aP[DST], bP[DST];
#pragma unroll
    for (int n = 0; n < DST; ++n) { aP[n] = 1.f; bP[n] = 0.f; }
    for (int i = 0; i < CHUNK; ++i) {
        int l = l0 + i;
        float xv = xsrow[l];
        float dr = dtb;
#pragma unroll
        for (int r = 0; r < DTR; ++r) dr += wdt[r] * xdb[(size_t)r * LT + l];
        float delta = softplus_f(dr);
#pragma unroll
        for (int n = 0; n < DST; ++n) {
            float dA  = __expf(delta * An[n]);
            float dBu = delta * xdb[(size_t)(DTR + n) * LT + l] * xv;
            aP[n] = aP[n] * dA;
            bP[n] = dA * bP[n] + dBu;
        }
    }

    // ---- cross-thread inclusive scan of transform pairs ----
#pragma unroll
    for (int n = 0; n < DST; ++n) { sA[t * DST + n] = aP[n]; sB[t * DST + n] = bP[n]; }
    __syncthreads();
    for (int off = 1; off < SCAN_T; off <<= 1) {
        float ta[DST], tb[DST];
        bool valid = (t >= off);
#pragma unroll
        for (int n = 0; n < DST; ++n) {
            ta[n] = valid ? sA[(t - off) * DST + n] : 1.f;
            tb[n] = valid ? sB[(t - off) * DST + n] : 0.f;
        }
        __syncthreads();
#pragma unroll
        for (int n = 0; n < DST; ++n) {
            float a2 = sA[t * DST + n], b2 = sB[t * DST + n];
            sA[t * DST + n] = ta[n] * a2;          // (a1*a2,
            sB[t * DST + n] = a2 * tb[n] + b2;     //  a2*b1 + b2)
        }
        __syncthreads();
    }

    // exclusive prefix -> incoming hidden state (h_init = 0 so only b matters)
    float h[DST];
#pragma unroll
    for (int n = 0; n < DST; ++n)
        h[n] = (t > 0) ? sB[(t - 1) * DST + n] : 0.f;

    // ---- pass 2: replay chunk with prefix, contract with Cs ----
    for (int i = 0; i < CHUNK; ++i) {
        int l = l0 + i;
        float xv = xsrow[l];
        float dr = dtb;
#pragma unroll
        for (int r = 0; r < DTR; ++r) dr += wdt[r] * xdb[(size_t)r * LT + l];
        float delta = softplus_f(dr);
        float yv = Dd * xv;
#pragma unroll
        for (int n = 0; n < DST; ++n) {
            float dA  = __expf(delta * An[n]);
            float dBu = delta * xdb[(size_t)(DTR + n) * LT + l] * xv;
            h[n] = dA * h[n] + dBu;
            yv += h[n] * xdb[(size_t)(DTR + DST + n) * LT + l];
        }
        y[(size_t)(b * DM + d) * LT + l] = yv;
    }
}

// =====================================================================
// Kernel 6: mean over 8 dirs + LayerNorm(channel) + silu(z) gate
// One wave per (b, l); lane covers channels {lane, lane+32, lane+64}.
// =====================================================================
__global__ __launch_bounds__(256) void k_ln_gate(const float* __restrict__ y,
                                                 const float* __restrict__ xz,
                                                 const float* __restrict__ lnw,
                                                 const float* __restrict__ lnb,
                                                 float* __restrict__ g) {
    int wave = (blockIdx.x * blockDim.x + threadIdx.x) >> 5;  // 3456 waves
    int lane = threadIdx.x & 31;
    int b = wave / L_SP, l = wave % L_SP;

    float v[3]; float sum = 0.f, sq = 0.f;
#pragma unroll
    for (int j = 0; j < 3; ++j) {
        int d = lane + 32 * j;
        float acc = 0.f;
#pragma unroll
        for (int k = 0; k < NDIR; ++k)
            acc += y[(size_t)(b * DM + d) * LT + k * L_SP + l];
        acc *= 0.125f;
        v[j] = acc; sum += acc; sq += acc * acc;
    }
#pragma unroll
    for (int off = 16; off >= 1; off >>= 1) {
        sum += __shfl_xor(sum, off, 32);
        sq  += __shfl_xor(sq,  off, 32);
    }
    float mu  = sum * (1.f / DM);
    float var = sq * (1.f / DM) - mu * mu;
    float inv = rsqrtf(var + 1e-5f);
#pragma unroll
    for (int j = 0; j < 3; ++j) {
        int d = lane + 32 * j;
        float zz = xz[(size_t)(b * L_SP + l) * (2 * DM) + DM + d];
        g[(size_t)(b * L_SP + l) * DM + d] =
            ((v[j] - mu) * inv * lnw[d] + lnb[d]) * silu_f(zz);
    }
}

// =====================================================================
// Kernel 7: out = g @ out_proj_w.T   [3456 x 96] * [96 x 96]
// =====================================================================
__global__ __launch_bounds__(128) void k_out_proj(const float* __restrict__ g,
                                                  const float* __restrict__ w,
                                                  float* __restrict__ out) {
    int wave = (blockIdx.x * blockDim.x + threadIdx.x) >> 5;  // 1296 waves
    int lane = threadIdx.x & 31;
    int mT = wave / 6, nT = wave % 6;
    int m0 = mT * 16, n0 = nT * 16;
    int half = lane >> 4, mn = lane & 15;

    const float* arow = g + (size_t)(m0 + mn) * DM;
    const float* brow = w + (size_t)(n0 + mn) * DM;
    v8f acc = {};
#pragma unroll
    for (int k0 = 0; k0 < DM; k0 += 4) {
        int k = k0 + 2 * half;
        v2f a; a.x = arow[k]; a.y = arow[k + 1];
        v2f b; b.x = brow[k]; b.y = brow[k + 1];
        acc = __builtin_amdgcn_wmma_f32_16x16x4_f32(false, a, false, b,
                                                    (short)0, acc, false, false);
    }
#pragma unroll
    for (int r = 0; r < 8; ++r)
        out[(size_t)(m0 + r + 8 * half) * DM + n0 + mn] = acc[r];
}

// =====================================================================
// Host launcher
// =====================================================================
extern "C" void kernel_launch(void* const* d_in, const int* in_sizes, int n_in,
                              void* d_out, int out_size, void* d_ws, size_t ws_size,
                              hipStream_t stream) {
    const float* x      = (const float*)d_in[0];
    const float* ipw    = (const float*)d_in[1];
    const float* cw     = (const float*)d_in[2];
    const float* cb     = (const float*)d_in[3];
    const float* xpw    = (const float*)d_in[4];
    const float* dtw    = (const float*)d_in[5];
    const float* dtb    = (const float*)d_in[6];
    const float* alogs  = (const float*)d_in[7];
    const float* ds     = (const float*)d_in[8];
    const float* lnw    = (const float*)d_in[9];
    const float* lnb    = (const float*)d_in[10];
    const float* opw    = (const float*)d_in[11];
    float* out = (float*)d_out;

    // workspace layout (floats)
    float* ws   = (float*)d_ws;
    float* xz   = ws;                                   // 3456*192   = 663552
    float* xc   = xz   + (size_t)B_SZ * L_SP * 2 * DM;  // 2*96*1728  = 331776
    float* xs   = xc   + (size_t)B_SZ * DM * L_SP;      // 2*96*13824 = 2654208
    float* xdbl = xs   + (size_t)B_SZ * DM * LT;        // 2*38*13824 = 1050624
    float* ybuf = xdbl + (size_t)B_SZ * XDR * LT;       // 2*96*13824 = 2654208
    float* gbuf = ybuf + (size_t)B_SZ * DM * LT;        // 3456*96    = 331776

    // 1) in_proj: 2592 waves / 4 per block
    k_in_proj<<<648, 128, 0, stream>>>(x, ipw, xz);
    // 2) depthwise conv + SiLU: 331776 threads
    k_conv_silu<<<1296, 256, 0, stream>>>(xz, cw, cb, xc);
    // 3) 8-direction gather: 2654208 threads
    k_dirs<<<10368, 256, 0, stream>>>(xc, xs);
    // 4) x_proj: 5184 waves / 4 per block
    k_x_proj<<<1296, 128, 0, stream>>>(xs, xpw, xdbl);
    // 5) selective scan: one WG per (b,d)
    k_scan<<<B_SZ * DM, SCAN_T, 0, stream>>>(xs, xdbl, dtw, dtb, alogs, ds, ybuf);
    // 6) mean + LN + gate: 3456 waves / 8 per block
    k_ln_gate<<<432, 256, 0, stream>>>(ybuf, xz, lnw, lnb, gbuf);
    // 7) out_proj: 1296 waves / 4 per block
    k_out_proj<<<324, 128, 0, stream>>>(gbuf, opw, out);
}